// GCNConvolution_81140522156078
// MI455X (gfx1250) — compile-verified
//
#include <hip/hip_runtime.h>

#define F_IN 165
#define HID  128
#define NOUT 2

typedef __attribute__((ext_vector_type(2))) float v2f;
typedef __attribute__((ext_vector_type(8))) float v8f;

// Guaranteed single-instruction, non-returning fp32 atomic add at device scope.
// RMW executes at the L2 atomic units; no return data -> no LOADcnt dependency.
__device__ __forceinline__ void atom_add(float* p, float v) {
  asm volatile("global_atomic_add_f32 %0, %1, off scope:SCOPE_DEV"
               :: "v"(p), "v"(v) : "memory");
}

// ---------------------------------------------------------------------------
// fp32 WMMA GEMM: C[M,NP] = A[M,K] * B[Kp,NP]   (NP, K compile-time)
// Preconditions: M % 16 == 0, NP % 16 == 0, B padded to Kp = ceil4(K) rows
// (rows K..Kp-1 zero). One wave per 16x16 C tile, V_WMMA_F32_16X16X4_F32.
// A 16x4 lane layout (ISA 7.12.2): lane l: row = l&15, comp j -> K = 2*(l>>4)+j
// Main K loop is branch-free; ragged tail uses clamped-address + select.
// ---------------------------------------------------------------------------
template <int NP, int K>
__global__ __launch_bounds__(256) void wmma_gemm_f32(const float* __restrict__ A,
                                                     const float* __restrict__ B,
                                                     float* __restrict__ C,
                                                     int M) {
  const int lane   = threadIdx.x & 31;
  const int wave   = threadIdx.x >> 5;
  const int nTiles = NP >> 4;
  const int tile   = blockIdx.x * 8 + wave;
  const int mTile  = tile / nTiles;
  const int nTile  = tile - mTile * nTiles;
  if (mTile * 16 >= M) return;  // wave-uniform: EXEC all-1s at every WMMA

  const int    half = lane >> 4;   // 0 or 1
  const int    l16  = lane & 15;
  const float* Ar   = A + (size_t)(mTile * 16 + l16) * K + 2 * half;  // lane's A ptr
  const float* Bc   = B + nTile * 16 + l16 + (size_t)(2 * half) * NP; // lane's B ptr

  constexpr int KMAIN = K & ~3;
  v8f acc = {};

#pragma unroll 4
  for (int kk = 0; kk < KMAIN; kk += 4) {
    v2f a = *reinterpret_cast<const v2f*>(Ar + kk);        // global_load_b64
    v2f b;
    b.x = Bc[(size_t)(kk + 0) * NP];
    b.y = Bc[(size_t)(kk + 1) * NP];
    acc = __builtin_amdgcn_wmma_f32_16x16x4_f32(false, a, false, b,
                                                (short)0, acc, false, false);
  }
  if constexpr (KMAIN < K) {  // ragged K tail: clamp address, select value
    const int k0 = KMAIN + 2 * half;
    const int c0 = (k0 + 0 < K) ? 2 * half + 0 : K - 1 - KMAIN;
    const int c1 = (k0 + 1 < K) ? 2 * half + 1 : K - 1 - KMAIN;
    v2f a, b;
    a.x = (k0 + 0 < K) ? Ar[KMAIN - 2 * half + c0] : 0.0f;
    a.y = (k0 + 1 < K) ? Ar[KMAIN - 2 * half + c1] : 0.0f;
    b.x = Bc[(size_t)(KMAIN + 0) * NP];  // padded B rows are zero
    b.y = Bc[(size_t)(KMAIN + 1) * NP];
    acc = __builtin_amdgcn_wmma_f32_16x16x4_f32(false, a, false, b,
                                                (short)0, acc, false, false);
  }

  float* Cc = C + nTile * 16 + l16;
#pragma unroll
  for (int j = 0; j < 8; ++j)  // C: VGPR j -> rows j (lanes 0-15) / j+8 (16-31)
    Cc[(size_t)(mTile * 16 + j + 8 * half) * NP] = acc[j];
}

// Pad/copy: dst[Hd,Wd] <- src[Hs,Ws], zero-filled outside.
__global__ __launch_bounds__(256) void pad_copy(const float* __restrict__ src,
                                                float* __restrict__ dst,
                                                int Hs, int Ws, int Wd, int totalDst) {
  int i = blockIdx.x * 256 + threadIdx.x;
  if (i >= totalDst) return;
  int r = i / Wd, c = i - r * Wd;
  dst[i] = (r < Hs && c < Ws) ? src[(size_t)r * Ws + c] : 0.0f;
}

// --- degree / normalization --------------------------------------------------
__global__ __launch_bounds__(256) void deg_init(float* deg, int n) {
  int i = blockIdx.x * 256 + threadIdx.x;
  if (i < n) deg[i] = 1.0f;  // self-loop
}

__global__ __launch_bounds__(256) void deg_count(const long long* __restrict__ dst,
                                                 float* deg, int e) {
  int i = blockIdx.x * 256 + threadIdx.x;
  if (i < e) atom_add(&deg[(int)dst[i]], 1.0f);
}

__global__ __launch_bounds__(256) void deg_rsqrt(float* deg, int n) {
  int i = blockIdx.x * 256 + threadIdx.x;
  if (i < n) deg[i] = rsqrtf(deg[i]);  // deg >= 1 always (self-loops)
}

// --- layer 1 aggregation (HID = 128 floats per node) -------------------------
// Seed accumulator with self-loop term: H[i] = dinv_i^2 * C1[i]
__global__ __launch_bounds__(256) void selfloop_l1(const float* __restrict__ C1,
                                                   const float* __restrict__ dinv,
                                                   float* __restrict__ H, int n) {
  int t    = blockIdx.x * 256 + threadIdx.x;  // n*32 threads, float4 each
  int node = t >> 5;
  int lane = t & 31;
  if (node >= n) return;
  float di = dinv[node];
  float w  = di * di;
  const float4 v = *reinterpret_cast<const float4*>(C1 + (size_t)node * HID + lane * 4);
  float4 o = make_float4(v.x * w, v.y * w, v.z * w, v.w * w);
  *reinterpret_cast<float4*>(H + (size_t)node * HID + lane * 4) = o;
}

// One wave per edge: lanes cover the 128-float row as float4 gathers + atomics.
// Working sets (h 51MB, accumulator 51MB, edges 51MB) all fit the 192MB L2, so
// the ~1.6GB of gather traffic and 410M fp32 atomics stay L2-resident.
__global__ __launch_bounds__(256) void scatter_l1(const long long* __restrict__ src,
                                                  const long long* __restrict__ dst,
                                                  const float* __restrict__ dinv,
                                                  const float* __restrict__ C1,
                                                  float* __restrict__ H, int e) {
  int t    = blockIdx.x * 256 + threadIdx.x;
  int edge = t >> 5;
  int lane = t & 31;
  if (edge >= e) return;
  int s = (int)src[edge];
  int d = (int)dst[edge];
  float norm = dinv[s] * dinv[d];
  const float4 v = *reinterpret_cast<const float4*>(C1 + (size_t)s * HID + lane * 4);
  float* o = H + (size_t)d * HID + lane * 4;
  atom_add(o + 0, v.x * norm);
  atom_add(o + 1, v.y * norm);
  atom_add(o + 2, v.z * norm);
  atom_add(o + 3, v.w * norm);
}

__global__ __launch_bounds__(256) void bias_relu(float* __restrict__ H,
                                                 const float* __restrict__ b,
                                                 int total) {
  int i = blockIdx.x * 256 + threadIdx.x;
  if (i < total) {
    int f = i & (HID - 1);
    H[i] = fmaxf(H[i] + b[f], 0.0f);
  }
}

// --- layer 2 aggregation (only NOUT = 2 floats per node) ---------------------
// T is [n,16] (padded GEMM output); cols 0..1 are real.
__global__ __launch_bounds__(256) void out_init(const float* __restrict__ T,
                                                const float* __restrict__ dinv,
                                                const float* __restrict__ b2,
                                                float* __restrict__ out, int n) {
  int i = blockIdx.x * 256 + threadIdx.x;
  if (i >= n) return;
  float di = dinv[i];
  float w  = di * di;
  out[(size_t)i * NOUT + 0] = T[(size_t)i * 16 + 0] * w + b2[0];
  out[(size_t)i * NOUT + 1] = T[(size_t)i * 16 + 1] * w + b2[1];
}

__global__ __launch_bounds__(256) void scatter_l2(const long long* __restrict__ src,
                                                  const long long* __restrict__ dst,
                                                  const float* __restrict__ dinv,
                                                  const float* __restrict__ T,
                                                  float* __restrict__ out, int e) {
  int i = blockIdx.x * 256 + threadIdx.x;
  if (i >= e) return;
  int s = (int)src[i];
  int d = (int)dst[i];
  float norm = dinv[s] * dinv[d];
  atom_add(out + (size_t)d * NOUT + 0, T[(size_t)s * 16 + 0] * norm);
  atom_add(out + (size_t)d * NOUT + 1, T[(size_t)s * 16 + 1] * norm);
}

extern "C" void kernel_launch(void* const* d_in, const int* in_sizes, int n_in,
                              void* d_out, int out_size, void* d_ws, size_t ws_size,
                              hipStream_t stream) {
  const float*     x   = (const float*)d_in[0];
  const long long* ei  = (const long long*)d_in[1];  // int64 [2, E]
  const float*     W1  = (const float*)d_in[2];
  const float*     b1  = (const float*)d_in[3];
  const float*     W2  = (const float*)d_in[4];
  const float*     b2  = (const float*)d_in[5];
  float*           out = (float*)d_out;

  const int Nn = in_sizes[0] / F_IN;   // 100000 (multiple of 16)
  const int E  = in_sizes[1] / 2;      // 3200000
  const long long* srcIdx = ei;
  const long long* dstIdx = ei + E;

  const int KP1 = (F_IN + 3) & ~3;     // 168
  const int TN  = 16;                  // padded layer-2 output width

  // workspace carve-out (256B aligned)
  char*  ws  = (char*)d_ws;
  size_t off = 0;
  auto carve = [&](size_t bytes) -> float* {
    float* p = (float*)(ws + off);
    off += (bytes + 255) & ~(size_t)255;
    return p;
  };
  float* dinv = carve((size_t)Nn * sizeof(float));
  float* W1p  = carve((size_t)KP1 * HID * sizeof(float));   // [168,128]
  float* W2p  = carve((size_t)HID * TN * sizeof(float));    // [128,16]
  float* C1   = carve((size_t)Nn * HID * sizeof(float));
  float* H    = carve((size_t)Nn * HID * sizeof(float));
  float* T    = carve((size_t)Nn * TN * sizeof(float));
  (void)ws_size; (void)n_in; (void)out_size;

  const int nodeBlk = (Nn + 255) / 256;
  const int edgeBlk = (E + 255) / 256;
  const int node32  = (int)(((size_t)Nn * 32 + 255) / 256);
  const int edge32  = (int)(((size_t)E * 32 + 255) / 256);
  const int hidBlk  = (int)(((size_t)Nn * HID + 255) / 256);

  // pad weights (B matrices) so GEMM inner loops are branch-free
  {
    int t1 = KP1 * HID;
    pad_copy<<<(t1 + 255) / 256, 256, 0, stream>>>(W1, W1p, F_IN, HID, HID, t1);
    int t2 = HID * TN;
    pad_copy<<<(t2 + 255) / 256, 256, 0, stream>>>(W2, W2p, HID, NOUT, TN, t2);
  }

  // normalization
  deg_init <<<nodeBlk, 256, 0, stream>>>(dinv, Nn);
  deg_count<<<edgeBlk, 256, 0, stream>>>(dstIdx, dinv, E);
  deg_rsqrt<<<nodeBlk, 256, 0, stream>>>(dinv, Nn);

  // layer 1: GEMM (WMMA) then aggregate in 128 dims
  {
    int grid = ((Nn / 16) * (HID / 16) + 7) / 8;
    wmma_gemm_f32<HID, F_IN><<<grid, 256, 0, stream>>>(x, W1p, C1, Nn);
  }
  selfloop_l1<<<node32, 256, 0, stream>>>(C1, dinv, H, Nn);
  scatter_l1 <<<edge32, 256, 0, stream>>>(srcIdx, dstIdx, dinv, C1, H, E);
  bias_relu  <<<hidBlk, 256, 0, stream>>>(H, b1, Nn * HID);

  // layer 2: GEMM (WMMA, padded N=16) then aggregate in 2 dims
  {
    int grid = ((Nn / 16) * (TN / 16) + 7) / 8;
    wmma_gemm_f32<TN, HID><<<grid, 256, 0, stream>>>(H, W2p, T, Nn);
  }
  out_init  <<<nodeBlk, 256, 0, stream>>>(T, dinv, b2, out, Nn);
  scatter_l2<<<edgeBlk, 256, 0, stream>>>(srcIdx, dstIdx, dinv, T, out, E);
}